// GraphStabilizer_68805376082238
// MI455X (gfx1250) — compile-verified
//
#include <hip/hip_runtime.h>

typedef __attribute__((ext_vector_type(2))) float v2f;
typedef __attribute__((ext_vector_type(4))) float v4f;
typedef __attribute__((ext_vector_type(8))) float v8f;

#define C_DIM 128
#define N_PTS 8192
#define B_SZ  4
#define K_NB  16
#define BN    (B_SZ * N_PTS)

__device__ __forceinline__ v8f wmma4(v2f a, v2f b, v8f c) {
  // V_WMMA_F32_16X16X4_F32: D = A(16x4) * B(4x16) + C(16x16), fp32 exact
  return __builtin_amdgcn_wmma_f32_16x16x4_f32(false, a, false, b, (short)0, c,
                                               false, false);
}

__device__ __forceinline__ v8f splat8(float x) {
  v8f v = {x, x, x, x, x, x, x, x};
  return v;
}

// ---------------------------------------------------------------------------
// Kernel A: offset MLP + center_xyz.
// One wave = 16 points. hidden = relu(feat @ W1^T + b1); delta = hidden @ W2^T + b2
// center = xyz + delta (3 valid output columns).
// LDS per wave: hidden tile 16x132 floats.
// ---------------------------------------------------------------------------
__global__ void __launch_bounds__(256) offset_kernel(
    const float* __restrict__ feat, const float* __restrict__ xyz,
    const float* __restrict__ W1, const float* __restrict__ b1,
    const float* __restrict__ W2, const float* __restrict__ b2,
    float* __restrict__ center) {
  extern __shared__ float smem[];
  const int tid  = threadIdx.x;
  const int wave = tid >> 5;
  const int lane = tid & 31;
  const int ln   = lane & 15;   // M (A-frag) / N (B,C-frag) index
  const int half = lane >> 4;   // selects K pair / upper 8 rows of C
  float* hidden  = smem + wave * (16 * 132);
  const int p0   = (blockIdx.x * 8 + wave) * 16;

  // ---- layer 1: 16x128 @ 128x128 ----
  const float* arow = feat + (size_t)(p0 + ln) * C_DIM + 2 * half;
  for (int nt = 0; nt < 8; ++nt) {
    v8f acc = splat8(b1[nt * 16 + ln]);
    const float* brow = W1 + (size_t)(nt * 16 + ln) * C_DIM + 2 * half;
    for (int kk = 0; kk < 128; kk += 4)
      acc = wmma4(*(const v2f*)(arow + kk), *(const v2f*)(brow + kk), acc);
#pragma unroll
    for (int r = 0; r < 8; ++r)
      hidden[(r + 8 * half) * 132 + nt * 16 + ln] = fmaxf(acc[r], 0.f);
  }
  __syncthreads();

  // ---- layer 2: 16x128 @ 128x3 (padded to 16 cols, scale kills cols >= 3) ----
  v8f acc = splat8((ln < 3) ? b2[ln] : 0.f);
  const float bscale = (ln < 3) ? 1.f : 0.f;
  const float* hrow = hidden + ln * 132 + 2 * half;
  const float* brow = W2 + (size_t)((ln < 3) ? ln : 0) * C_DIM + 2 * half;
  for (int kk = 0; kk < 128; kk += 4) {
    v2f b = *(const v2f*)(brow + kk);
    b.x *= bscale;
    b.y *= bscale;
    acc = wmma4(*(const v2f*)(hrow + kk), b, acc);
  }
  if (ln < 3) {
#pragma unroll
    for (int r = 0; r < 8; ++r) {
      const int p = p0 + r + 8 * half;
      center[(size_t)p * 3 + ln] = xyz[(size_t)p * 3 + ln] + acc[r];
    }
  }
}

// ---------------------------------------------------------------------------
// Kernel B: edge MLP + max-aggregation.
// One wave = one point; its K=16 neighbors form the M=16 tile.
// edge_feat = relu([rel_pos|nb_feat](16x131) @ W1^T + b1) @ W2^T + b2
// agg = max over the 16 rows.
// LDS: W1 (128x132) + W2 (128x132) staged once per block; per wave
// input tile 16x132 + hidden tile 16x132. Total 270336 B.
// ---------------------------------------------------------------------------
__global__ void __launch_bounds__(256) edge_kernel(
    const float* __restrict__ feat, const float* __restrict__ xyz,
    const int* __restrict__ knn, const float* __restrict__ center,
    const float* __restrict__ W1, const float* __restrict__ b1,
    const float* __restrict__ W2, const float* __restrict__ b2,
    float* __restrict__ agg) {
  extern __shared__ float smem[];
  float* W1s = smem;             // 128*132, col 131 zero-padded
  float* W2s = smem + 128 * 132; // 128*132 (cols >=128 unused)
  const int tid  = threadIdx.x;
  const int wave = tid >> 5;
  const int lane = tid & 31;
  const int ln   = lane & 15;
  const int half = lane >> 4;
  float* tileA  = smem + 2 * 128 * 132 + wave * (2 * 16 * 132);
  float* hidden = tileA + 16 * 132;

  // Stage weights (shared by all 64 points of this block).
  for (int i = tid; i < 128 * 132; i += 256) {
    const int r = i / 132, c = i - r * 132;
    W1s[i] = (c < 131) ? W1[(size_t)r * 131 + c] : 0.f;
    W2s[i] = (c < 128) ? W2[(size_t)r * 128 + c] : 0.f;
  }
  __syncthreads();

  const int base = blockIdx.x * 64 + wave * 8;
  for (int pt = 0; pt < 8; ++pt) {
    const int p = base + pt;
    const int b = p >> 13;  // N = 8192

    // ---- stage edge_input tile: row = neighbor, cols [rel(3) | feat(128) | pad] ----
    const int idx = knn[(size_t)p * K_NB + ln];
    const int nb  = (b << 13) + idx;
    {
      const float* src = feat + (size_t)nb * C_DIM + half * 64;
      float* drow = tileA + ln * 132 + 3 + half * 64;
#pragma unroll
      for (int j = 0; j < 64; j += 4) {
        const v4f v = *(const v4f*)(src + j);
        drow[j + 0] = v.x;
        drow[j + 1] = v.y;
        drow[j + 2] = v.z;
        drow[j + 3] = v.w;
      }
    }
    if (half == 0) {
#pragma unroll
      for (int c = 0; c < 3; ++c)
        tileA[ln * 132 + c] = xyz[(size_t)nb * 3 + c] - center[(size_t)p * 3 + c];
    } else {
      tileA[ln * 132 + 131] = 0.f;
    }
    __syncthreads();

    // ---- layer 1: 16x132(pad) @ 132x128 ----
    const float* ar = tileA + ln * 132 + 2 * half;
    for (int nt = 0; nt < 8; ++nt) {
      v8f acc = splat8(b1[nt * 16 + ln]);
      const float* br = W1s + (nt * 16 + ln) * 132 + 2 * half;
      for (int kk = 0; kk < 132; kk += 4)
        acc = wmma4(*(const v2f*)(ar + kk), *(const v2f*)(br + kk), acc);
#pragma unroll
      for (int r = 0; r < 8; ++r)
        hidden[(r + 8 * half) * 132 + nt * 16 + ln] = fmaxf(acc[r], 0.f);
    }
    __syncthreads();

    // ---- layer 2 + max over the 16 neighbor rows ----
    const float* hr = hidden + ln * 132 + 2 * half;
    for (int nt = 0; nt < 8; ++nt) {
      v8f acc = splat8(b2[nt * 16 + ln]);
      const float* br = W2s + (nt * 16 + ln) * 132 + 2 * half;
      for (int kk = 0; kk < 128; kk += 4)
        acc = wmma4(*(const v2f*)(hr + kk), *(const v2f*)(br + kk), acc);
      float m = acc[0];
#pragma unroll
      for (int r = 1; r < 8; ++r) m = fmaxf(m, acc[r]);
      m = fmaxf(m, __shfl_xor(m, 16, 32));  // combine row halves (M and M+8)
      if (half == 0) agg[(size_t)p * C_DIM + nt * 16 + ln] = m;
    }
    __syncthreads();
  }
}

// ---------------------------------------------------------------------------
// Kernel C: update MLP (+ residual).
// One wave = 16 points. upd_in = [agg | feat] (16x256);
// feat_out = feat + relu(upd_in @ W1^T + b1) @ W2^T + b2.
// LDS per wave: input tile 16x260 + hidden tile 16x132.
// In-place safe when featOut == featIn (each block only touches its own rows,
// reads complete before writes).
// ---------------------------------------------------------------------------
__global__ void __launch_bounds__(256) update_kernel(
    const float* __restrict__ featIn, const float* __restrict__ agg,
    const float* __restrict__ W1, const float* __restrict__ b1,
    const float* __restrict__ W2, const float* __restrict__ b2,
    float* __restrict__ featOut) {
  extern __shared__ float smem[];
  const int tid  = threadIdx.x;
  const int wave = tid >> 5;
  const int lane = tid & 31;
  const int ln   = lane & 15;
  const int half = lane >> 4;
  float* tileA  = smem + wave * (16 * 260 + 16 * 132);
  float* hidden = tileA + 16 * 260;
  const int p0  = (blockIdx.x * 8 + wave) * 16;

  // stage [agg | feat] 16x256, stride 260 for bank spread
  {
    const float* src = (half ? featIn : agg) + (size_t)(p0 + ln) * C_DIM;
    float* drow = tileA + ln * 260 + half * 128;
#pragma unroll
    for (int j = 0; j < 128; j += 4) *(v4f*)(drow + j) = *(const v4f*)(src + j);
  }
  __syncthreads();

  // ---- layer 1: 16x256 @ 256x128 (B-frags straight from L2-resident weights) ----
  const float* ar = tileA + ln * 260 + 2 * half;
  for (int nt = 0; nt < 8; ++nt) {
    v8f acc = splat8(b1[nt * 16 + ln]);
    const float* br = W1 + (size_t)(nt * 16 + ln) * 256 + 2 * half;
    for (int kk = 0; kk < 256; kk += 4)
      acc = wmma4(*(const v2f*)(ar + kk), *(const v2f*)(br + kk), acc);
#pragma unroll
    for (int r = 0; r < 8; ++r)
      hidden[(r + 8 * half) * 132 + nt * 16 + ln] = fmaxf(acc[r], 0.f);
  }
  __syncthreads();

  // ---- layer 2: 16x128 @ 128x128 + residual ----
  const float* hr = hidden + ln * 132 + 2 * half;
  for (int nt = 0; nt < 8; ++nt) {
    v8f acc = splat8(b2[nt * 16 + ln]);
    const float* br = W2 + (size_t)(nt * 16 + ln) * C_DIM + 2 * half;
    for (int kk = 0; kk < 128; kk += 4)
      acc = wmma4(*(const v2f*)(hr + kk), *(const v2f*)(br + kk), acc);
#pragma unroll
    for (int r = 0; r < 8; ++r) {
      const size_t o = (size_t)(p0 + r + 8 * half) * C_DIM + nt * 16 + ln;
      featOut[o] = featIn[o] + acc[r];
    }
  }
}

// ---------------------------------------------------------------------------
extern "C" void kernel_launch(void* const* d_in, const int* in_sizes, int n_in,
                              void* d_out, int out_size, void* d_ws,
                              size_t ws_size, hipStream_t stream) {
  const float* xyz     = (const float*)d_in[0];
  const float* feat0   = (const float*)d_in[1];
  const int*   knn     = (const int*)d_in[2];
  const float* off_W1  = (const float*)d_in[3];
  const float* off_b1  = (const float*)d_in[4];
  const float* off_W2  = (const float*)d_in[5];
  const float* off_b2  = (const float*)d_in[6];
  const float* edge_W1 = (const float*)d_in[7];
  const float* edge_b1 = (const float*)d_in[8];
  const float* edge_W2 = (const float*)d_in[9];
  const float* edge_b2 = (const float*)d_in[10];
  const float* upd_W1  = (const float*)d_in[11];
  const float* upd_b1  = (const float*)d_in[12];
  const float* upd_W2  = (const float*)d_in[13];
  const float* upd_b2  = (const float*)d_in[14];
  float* out = (float*)d_out;

  const int T = in_sizes[7] / (C_DIM * (3 + C_DIM));  // = 3

  // workspace: center_xyz (BN*3, padded to BN*4) + agg (BN*128)
  float* center = (float*)d_ws;
  float* agg    = center + (size_t)BN * 4;

  const int A_LDS = 8 * 16 * 132 * 4;                        //  67584 B
  const int B_LDS = (2 * 128 * 132 + 8 * 2 * 16 * 132) * 4;  // 270336 B
  const int C_LDS = 8 * (16 * 260 + 16 * 132) * 4;           // 200704 B
  hipFuncSetAttribute(reinterpret_cast<const void*>(offset_kernel),
                      hipFuncAttributeMaxDynamicSharedMemorySize, A_LDS);
  hipFuncSetAttribute(reinterpret_cast<const void*>(edge_kernel),
                      hipFuncAttributeMaxDynamicSharedMemorySize, B_LDS);
  hipFuncSetAttribute(reinterpret_cast<const void*>(update_kernel),
                      hipFuncAttributeMaxDynamicSharedMemorySize, C_LDS);

  for (int t = 0; t < T; ++t) {
    const float* Fin = (t == 0) ? feat0 : out;  // features ping-pong via d_out
    offset_kernel<<<BN / 128, 256, A_LDS, stream>>>(Fin, xyz, off_W1, off_b1,
                                                    off_W2, off_b2, center);
    edge_kernel<<<BN / 64, 256, B_LDS, stream>>>(
        Fin, xyz, knn, center, edge_W1 + (size_t)t * C_DIM * 131,
        edge_b1 + (size_t)t * C_DIM, edge_W2 + (size_t)t * C_DIM * C_DIM,
        edge_b2 + (size_t)t * C_DIM, agg);
    update_kernel<<<BN / 128, 256, C_LDS, stream>>>(
        Fin, agg, upd_W1 + (size_t)t * C_DIM * 256, upd_b1 + (size_t)t * C_DIM,
        upd_W2 + (size_t)t * C_DIM * C_DIM, upd_b2 + (size_t)t * C_DIM, out);
  }
  (void)n_in;
  (void)out_size;
  (void)ws_size;
}